// GraphMemory_28260884807794
// MI455X (gfx1250) — compile-verified
//
#include <hip/hip_runtime.h>

// Knowledge-graph embedding update: gather h/r/t rows, new = 0.9*h + 0.1*(h+r-t),
// scatter into a copy of `memory` at h_idx with last-write-wins duplicate semantics.
// Pure bandwidth-bound elementwise op: optimized for B128 vector memory, wave32
// row mapping (32 lanes x float4 == one 128-f32 row), NT stores, index prefetch.

#define EMB 128
#define EMB4 32 // float4 chunks per row

typedef __attribute__((ext_vector_type(4))) float f32x4;

// ---------------------------------------------------------------------------
// Kernel 1: out = memory (streaming copy, B128), winner[] = -1.
// NT store hint on the copy output so the 256 MB write stream does not evict
// the memory table from L2 (192 MB) -- the random gathers in kernel 3 hit L2
// much more often if the table stays resident.
// ---------------------------------------------------------------------------
__global__ __launch_bounds__(256) void k_copy_init(const f32x4* __restrict__ mem,
                                                   f32x4* __restrict__ out,
                                                   int* __restrict__ winner,
                                                   int n4, int n_ent) {
    int i = blockIdx.x * 256 + threadIdx.x;
    if (i < n4) {
        f32x4 v = mem[i];                       // regular-temporal: warm L2
        __builtin_nontemporal_store(v, &out[i]); // NT: don't pollute L2
    }
    if (i < n_ent) winner[i] = -1;
}

// ---------------------------------------------------------------------------
// Kernel 2: last-write-wins duplicate resolution.
// winner[h] = max triple index e with h_idx[e] == h.
// Streaming index read; global_prefetch_b8 on the upcoming index cachelines.
// ---------------------------------------------------------------------------
__global__ __launch_bounds__(256) void k_winner(const int* __restrict__ h_idx,
                                                int* __restrict__ winner,
                                                int n_tri) {
    int e = blockIdx.x * 256 + threadIdx.x;
    if (e < n_tri) {
        if (e + 4096 < n_tri)
            __builtin_prefetch(&h_idx[e + 4096], 0, 1); // -> global_prefetch_b8
        atomicMax(&winner[h_idx[e]], e);
    }
}

// ---------------------------------------------------------------------------
// Kernel 3: one wave32 per entity row (32 lanes x 16B = one 512B row per
// vector memory instruction). Uniform early-out for untouched rows; winning
// triple's r/t indices scalarized via readfirstlane so the row base addresses
// live in SGPRs. Compute matches the reference's exact op order:
//   msg = h + r - t ;  new = 0.9*h + 0.1*msg
// ---------------------------------------------------------------------------
__global__ __launch_bounds__(256) void k_apply(const f32x4* __restrict__ mem,
                                               const f32x4* __restrict__ rel,
                                               const int* __restrict__ r_idx,
                                               const int* __restrict__ t_idx,
                                               const int* __restrict__ winner,
                                               f32x4* __restrict__ out,
                                               int n_ent) {
    int gid  = blockIdx.x * 256 + threadIdx.x;
    int row  = gid >> 5;   // wave32: one wave per row
    int lane = gid & 31;
    if (row >= n_ent) return;

    int e = winner[row];   // wave-uniform (all lanes read same dword)
    if (e < 0) return;     // uniform -> whole-wave early exit
    e = __builtin_amdgcn_readfirstlane(e);

    int r = __builtin_amdgcn_readfirstlane(r_idx[e]);
    int t = __builtin_amdgcn_readfirstlane(t_idx[e]);

    f32x4 h  = mem[(long)row * EMB4 + lane];  // global_load_b128
    f32x4 rv = rel[(long)r   * EMB4 + lane];  // 1000x512B table: L2-hot
    f32x4 tv = mem[(long)t   * EMB4 + lane];

    f32x4 msg = h + rv - tv;
    f32x4 nv  = 0.9f * h + 0.1f * msg;

    __builtin_nontemporal_store(nv, &out[(long)row * EMB4 + lane]);
}

// ---------------------------------------------------------------------------
extern "C" void kernel_launch(void* const* d_in, const int* in_sizes, int n_in,
                              void* d_out, int out_size, void* d_ws, size_t ws_size,
                              hipStream_t stream) {
    const float* mem   = (const float*)d_in[0];
    const float* rel   = (const float*)d_in[1];
    const int*   h_idx = (const int*)d_in[2];
    const int*   r_idx = (const int*)d_in[3];
    const int*   t_idx = (const int*)d_in[4];
    float*       out   = (float*)d_out;
    int*         winner = (int*)d_ws;          // n_ent ints of scratch

    const int n_ent = in_sizes[0] / EMB;
    const int n_tri = in_sizes[2];
    const int n4    = in_sizes[0] / 4;         // float4 count of the table

    // 1) copy table to output + reset winner scratch (every call: deterministic)
    k_copy_init<<<(n4 + 255) / 256, 256, 0, stream>>>(
        (const f32x4*)mem, (f32x4*)out, winner, n4, n_ent);

    // 2) last-write-wins winner per entity
    k_winner<<<(n_tri + 255) / 256, 256, 0, stream>>>(h_idx, winner, n_tri);

    // 3) apply winning triple per touched row (one wave32 per row)
    long total = (long)n_ent * 32;
    k_apply<<<(int)((total + 255) / 256), 256, 0, stream>>>(
        (const f32x4*)mem, (const f32x4*)rel, r_idx, t_idx, winner, (f32x4*)out, n_ent);
}